// SpatialGcnMamba_24816321036305
// MI455X (gfx1250) — compile-verified
//
#include <hip/hip_runtime.h>
#include <math.h>

// ---------------------------------------------------------------------------
// Problem dims (compile-time, from the reference)
// ---------------------------------------------------------------------------
#define BB 2048
#define LL 17
#define DD 256
#define EE 512
#define NN 8
#define RR 16
#define HH 1024
#define MM (BB * LL)          // 34816 rows, multiple of 128

typedef __attribute__((ext_vector_type(2))) float v2f;
typedef __attribute__((ext_vector_type(8))) float v8f;

// 17-joint adjacency as bitmasks (from CONNECTIONS, symmetric)
__device__ __constant__ unsigned int kAdj[17] = {
    146u, 5u, 10u, 4u, 33u, 80u, 32u, 257u, 19072u, 1280u, 512u,
    4352u, 10240u, 4096u, 33024u, 81920u, 32768u
};

__device__ __forceinline__ float siluf(float x) {
    return x / (1.0f + expf(-x));
}
__device__ __forceinline__ float softplusf(float x) {
    return (x > 30.0f) ? x : log1pf(expf(x));
}
__device__ __forceinline__ float gelu_exact(float x) {
    return 0.5f * x * (1.0f + erff(x * 0.70710678118654752440f));
}

// ---------------------------------------------------------------------------
// fp32 WMMA GEMM:  C[M,N] = epilogue( A[M,K(lda)] @ W[N,K]^T )
// Block = 8 waves; wave w -> m-tile (16 rows); all waves share NT 16-col tiles.
// W k-chunks (KC) staged in LDS (double-buffered, +4-word row pad -> conflict
// free ds_load_b64 B-fragments).  Next chunk's W slab + A fragments are
// global-loaded into registers while the current chunk's WMMAs run.
// MODE: 0 none | 1 +bias | 2 +bias,softplus | 3 +bias,gelu
//       4 res + acc*scale[col] | 5 +bias + res
// Requires: M % 128 == 0, N == gridDim.y*NT*16, K % KC == 0.
// ---------------------------------------------------------------------------
template <int NT, int KC, int MODE>
__global__ void gemm_f32_wmma(const float* __restrict__ A,
                              const float* __restrict__ W,
                              const float* __restrict__ bias,
                              const float* __restrict__ res,
                              const float* __restrict__ scale,
                              float* __restrict__ C,
                              int M, int N, int K, int lda) {
    constexpr int LDW  = KC + 4;              // padded LDS row stride (words)
    constexpr int KC4  = KC / 4;              // k-steps per chunk
    constexpr int TOT4 = NT * 16 * KC / 4;    // float4s per W slab
    constexpr int PER  = TOT4 / 256;          // float4s per thread

    __shared__ __align__(16) float Wl[2][NT * 16 * LDW];

    const int tid  = threadIdx.x;
    const int lane = tid & 31;
    const int wave = tid >> 5;
    const int mt   = blockIdx.x * 8 + wave;   // m-tile (wave-uniform)
    const int m0   = mt * 16;
    const int nb   = blockIdx.y * (NT * 16);
    const int r    = lane & 15;               // A-row / B-col within tile
    const int hk   = lane >> 4;               // K half-select

    const float* Arow = A + (size_t)(m0 + r) * lda;
    const int NC = K / KC;

    // ---- prologue: load chunk 0 (W slab -> LDS, A frags -> regs) ----
    float4 wr[PER];
#pragma unroll
    for (int p = 0; p < PER; ++p) {
        const int idx4 = tid + 256 * p;
        const int col  = (idx4 * 4) / KC;
        const int kk   = (idx4 * 4) % KC;
        wr[p] = *(const float4*)(W + (size_t)(nb + col) * K + kk);
    }
    v2f areg[KC4];
#pragma unroll
    for (int j = 0; j < KC4; ++j)
        areg[j] = *(const v2f*)(Arow + 4 * j + 2 * hk);
#pragma unroll
    for (int p = 0; p < PER; ++p) {
        const int idx4 = tid + 256 * p;
        const int col  = (idx4 * 4) / KC;
        const int kk   = (idx4 * 4) % KC;
        *(float4*)(&Wl[0][col * LDW + kk]) = wr[p];
    }
    __syncthreads();

    v8f acc[NT];
#pragma unroll
    for (int t = 0; t < NT; ++t)
        acc[t] = (v8f){0.f, 0.f, 0.f, 0.f, 0.f, 0.f, 0.f, 0.f};

    for (int kc = 0; kc < NC; ++kc) {
        const bool more = (kc + 1) < NC;
        float4 wr2[PER];
        v2f an[KC4];
        if (more) {
            const int kb = (kc + 1) * KC;
#pragma unroll
            for (int p = 0; p < PER; ++p) {
                const int idx4 = tid + 256 * p;
                const int col  = (idx4 * 4) / KC;
                const int kk   = (idx4 * 4) % KC;
                wr2[p] = *(const float4*)(W + (size_t)(nb + col) * K + kb + kk);
            }
#pragma unroll
            for (int j = 0; j < KC4; ++j)
                an[j] = *(const v2f*)(Arow + kb + 4 * j + 2 * hk);
            if (kc + 2 < NC) {
                const int c0 = (tid * 4) / KC;
                __builtin_prefetch((const void*)(W + (size_t)(nb + c0) * K +
                                                 (kc + 2) * KC + (tid * 4) % KC), 0, 0);
            }
        }
        // ---- compute chunk kc from LDS (ds_load_b64 -> v_wmma) ----
        const float* Wb = &Wl[kc & 1][0];
#pragma unroll
        for (int j = 0; j < KC4; ++j) {
            const v2f a = areg[j];
#pragma unroll
            for (int t = 0; t < NT; ++t) {
                const v2f b = *(const v2f*)(Wb + (t * 16 + r) * LDW + 4 * j + 2 * hk);
                acc[t] = __builtin_amdgcn_wmma_f32_16x16x4_f32(
                    false, a, false, b, (short)0, acc[t], false, false);
            }
        }
        if (more) {
            float* dst = &Wl[(kc + 1) & 1][0];
#pragma unroll
            for (int p = 0; p < PER; ++p) {
                const int idx4 = tid + 256 * p;
                const int col  = (idx4 * 4) / KC;
                const int kk   = (idx4 * 4) % KC;
                *(float4*)(dst + col * LDW + kk) = wr2[p];
            }
#pragma unroll
            for (int j = 0; j < KC4; ++j) areg[j] = an[j];
        }
        __syncthreads();
    }

    // ---- epilogue ----
#pragma unroll
    for (int t = 0; t < NT; ++t) {
        const int col = nb + t * 16 + r;
#pragma unroll
        for (int i = 0; i < 8; ++i) {
            const int row = m0 + i + 8 * hk;
            const size_t idx = (size_t)row * N + col;
            float v = acc[t][i];
            if (MODE == 1) v += bias[col];
            if (MODE == 2) v = softplusf(v + bias[col]);
            if (MODE == 3) v = gelu_exact(v + bias[col]);
            if (MODE == 4) v = res[idx] + v * scale[col];
            if (MODE == 5) v = res[idx] + v + bias[col];
            C[idx] = v;
        }
    }
}

// ---------------------------------------------------------------------------
// Dual LayerNorm: one wave per row, produces LN(x)*g1+b1 and LN(x)*g2+b2
// ---------------------------------------------------------------------------
__global__ void ln_dual_kernel(const float* __restrict__ x,
                               const float* __restrict__ g1, const float* __restrict__ b1,
                               const float* __restrict__ g2, const float* __restrict__ b2,
                               float* __restrict__ o1, float* __restrict__ o2) {
    const int lane = threadIdx.x & 31;
    const int wave = threadIdx.x >> 5;
    const size_t m = (size_t)blockIdx.x * (blockDim.x >> 5) + wave;
    const float* row = x + m * DD;
    float v[8], s = 0.f, s2 = 0.f;
#pragma unroll
    for (int i = 0; i < 8; ++i) {
        float t = row[lane + 32 * i];
        v[i] = t; s += t; s2 += t * t;
    }
#pragma unroll
    for (int off = 16; off; off >>= 1) {
        s  += __shfl_xor(s, off);
        s2 += __shfl_xor(s2, off);
    }
    const float mu = s * (1.0f / DD);
    const float rs = rsqrtf(s2 * (1.0f / DD) - mu * mu + 1e-5f);
#pragma unroll
    for (int i = 0; i < 8; ++i) {
        const int d = lane + 32 * i;
        const float nh = (v[i] - mu) * rs;
        o1[m * DD + d] = nh * g1[d] + b1[d];
        o2[m * DD + d] = nh * g2[d] + b2[d];
    }
}

// Single LayerNorm
__global__ void ln_kernel(const float* __restrict__ x,
                          const float* __restrict__ g, const float* __restrict__ b,
                          float* __restrict__ o) {
    const int lane = threadIdx.x & 31;
    const int wave = threadIdx.x >> 5;
    const size_t m = (size_t)blockIdx.x * (blockDim.x >> 5) + wave;
    const float* row = x + m * DD;
    float v[8], s = 0.f, s2 = 0.f;
#pragma unroll
    for (int i = 0; i < 8; ++i) {
        float t = row[lane + 32 * i];
        v[i] = t; s += t; s2 += t * t;
    }
#pragma unroll
    for (int off = 16; off; off >>= 1) {
        s  += __shfl_xor(s, off);
        s2 += __shfl_xor(s2, off);
    }
    const float mu = s * (1.0f / DD);
    const float rs = rsqrtf(s2 * (1.0f / DD) - mu * mu + 1e-5f);
#pragma unroll
    for (int i = 0; i < 8; ++i) {
        const int d = lane + 32 * i;
        o[m * DD + d] = (v[i] - mu) * rs * g[d] + b[d];
    }
}

// ---------------------------------------------------------------------------
// GCN aggregate + BN + ReLU + residual:
//   x1 = x + relu( xn + BN_i( sum_j w_ij * hV[b,j,:] + hU[b,i,:] ) )
// ---------------------------------------------------------------------------
__global__ void gcn_fuse_kernel(const float* __restrict__ x,
                                const float* __restrict__ xn,
                                const float* __restrict__ hV,
                                const float* __restrict__ hU,
                                const float* __restrict__ bn_g, const float* __restrict__ bn_b,
                                const float* __restrict__ bn_mean, const float* __restrict__ bn_var,
                                float* __restrict__ x1) {
    const size_t tid = (size_t)blockIdx.x * blockDim.x + threadIdx.x; // M*D
    const size_t m = tid >> 8;
    const int d = tid & 255;
    const int i = (int)(m % LL);
    const size_t bbase = (m - i) * DD; // (b*L)*D
    const float di = rsqrtf((float)__popc(kAdj[i]));
    float agg = 0.f;
    unsigned int mm = kAdj[i];
    while (mm) {
        const int j = __ffs(mm) - 1;
        mm &= mm - 1;
        const float wij = di * rsqrtf((float)__popc(kAdj[j]));
        agg += wij * hV[bbase + (size_t)j * DD + d];
    }
    float v = agg + hU[m * DD + d];
    v = (v - bn_mean[i]) * rsqrtf(bn_var[i] + 1e-5f) * bn_g[i] + bn_b[i];
    const float r = xn[m * DD + d] + v;
    x1[m * DD + d] = x[m * DD + d] + (r > 0.f ? r : 0.f);
}

// ---------------------------------------------------------------------------
// Causal depthwise conv (K=4) + SiLU, with optional time reversal of input u.
// u lives as cols [0,E) of xz (row stride 2E).  Output xc in branch-time order.
// ---------------------------------------------------------------------------
__global__ void conv_silu_kernel(const float* __restrict__ xz,
                                 const float* __restrict__ w,
                                 const float* __restrict__ bias,
                                 float* __restrict__ xc, int rev) {
    const size_t tid = (size_t)blockIdx.x * blockDim.x + threadIdx.x; // M*E
    const size_t row = tid >> 9;
    const int e = tid & 511;
    const int t = (int)(row % LL);
    const size_t b17 = row - t; // b*L
    float acc = bias[e];
#pragma unroll
    for (int k = 0; k < 4; ++k) {
        const int tt = t - 3 + k;
        if (tt >= 0) {
            const int l = rev ? (LL - 1 - tt) : tt;
            acc += xz[(b17 + l) * (2 * EE) + e] * w[e * 4 + k];
        }
    }
    xc[row * EE + e] = siluf(acc);
}

// ---------------------------------------------------------------------------
// Selective scan: one thread per (b,e); N=8 state in registers, 17 steps.
// combine=0: ybuf = y_f ;  combine=1 (rev branch): ybuf = 0.5*(ybuf + y_b)
// ---------------------------------------------------------------------------
__global__ void ssm_scan_kernel(const float* __restrict__ dt,
                                const float* __restrict__ xc,
                                const float* __restrict__ xdbl,
                                const float* __restrict__ xz,     // z = cols [E,2E)
                                const float* __restrict__ A_log,
                                const float* __restrict__ Dskip,
                                float* __restrict__ ybuf,
                                int rev, int combine) {
    const size_t tid = (size_t)blockIdx.x * blockDim.x + threadIdx.x; // B*E
    const size_t b = tid >> 9;
    const int e = tid & 511;
    float Ae[NN], h[NN];
#pragma unroll
    for (int n = 0; n < NN; ++n) {
        Ae[n] = -expf(A_log[e * NN + n]);
        h[n] = 0.f;
    }
    const float Dsk = Dskip[e];
    for (int t = 0; t < LL; ++t) {
        const size_t rm = b * LL + t;
        const float dtv = dt[rm * EE + e];
        const float uv  = xc[rm * EE + e];
        const float du  = dtv * uv;
        const float* xd = xdbl + rm * (RR + 2 * NN);
        float y = 0.f;
#pragma unroll
        for (int n = 0; n < NN; ++n) {
            h[n] = expf(dtv * Ae[n]) * h[n] + du * xd[RR + n];
            y += h[n] * xd[RR + NN + n];
        }
        y += uv * Dsk;
        const int l = rev ? (LL - 1 - t) : t;
        const float zv = xz[(b * LL + l) * (2 * EE) + EE + e];
        y *= siluf(zv);
        const size_t oi = (b * LL + l) * EE + e;
        if (combine) ybuf[oi] = 0.5f * (ybuf[oi] + y);
        else         ybuf[oi] = y;
    }
}

// ---------------------------------------------------------------------------
// Token fusion: logits = [x1|x2] @ ts_w^T + ts_b ; softmax over 2 ; blend.
// One wave per row.
// ---------------------------------------------------------------------------
__global__ void ts_fuse_kernel(const float* __restrict__ x1,
                               const float* __restrict__ x2,
                               const float* __restrict__ tsw,
                               const float* __restrict__ tsb,
                               float* __restrict__ xf) {
    const int lane = threadIdx.x & 31;
    const int wave = threadIdx.x >> 5;
    const size_t m = (size_t)blockIdx.x * (blockDim.x >> 5) + wave;
    float a[8], c[8], l0 = 0.f, l1 = 0.f;
#pragma unroll
    for (int i = 0; i < 8; ++i) {
        const int d = lane + 32 * i;
        a[i] = x1[m * DD + d];
        c[i] = x2[m * DD + d];
        l0 += a[i] * tsw[d]          + c[i] * tsw[DD + d];
        l1 += a[i] * tsw[2 * DD + d] + c[i] * tsw[3 * DD + d];
    }
#pragma unroll
    for (int off = 16; off; off >>= 1) {
        l0 += __shfl_xor(l0, off);
        l1 += __shfl_xor(l1, off);
    }
    l0 += tsb[0];
    l1 += tsb[1];
    const float mx = fmaxf(l0, l1);
    const float e0 = expf(l0 - mx), e1 = expf(l1 - mx);
    const float inv = 1.0f / (e0 + e1);
    const float a0 = e0 * inv, a1 = e1 * inv;
#pragma unroll
    for (int i = 0; i < 8; ++i) {
        const int d = lane + 32 * i;
        xf[m * DD + d] = a[i] * a0 + c[i] * a1;
    }
}

// ---------------------------------------------------------------------------
// Host-side launch
// ---------------------------------------------------------------------------
extern "C" void kernel_launch(void* const* d_in, const int* in_sizes, int n_in,
                              void* d_out, int out_size, void* d_ws, size_t ws_size,
                              hipStream_t stream) {
    const float* x         = (const float*)d_in[0];
    const float* ln_gcn_g  = (const float*)d_in[1];
    const float* ln_gcn_b  = (const float*)d_in[2];
    const float* gcn_U_w   = (const float*)d_in[3];
    const float* gcn_U_b   = (const float*)d_in[4];
    const float* gcn_V_w   = (const float*)d_in[5];
    const float* gcn_V_b   = (const float*)d_in[6];
    const float* bn_g      = (const float*)d_in[7];
    const float* bn_b      = (const float*)d_in[8];
    const float* bn_mean   = (const float*)d_in[9];
    const float* bn_var    = (const float*)d_in[10];
    const float* ln1_g     = (const float*)d_in[11];
    const float* ln1_b     = (const float*)d_in[12];
    const float* in_proj_w = (const float*)d_in[13];
    const float* conv_w    = (const float*)d_in[14];
    const float* conv_b    = (const float*)d_in[15];
    const float* convb_w   = (const float*)d_in[16];
    const float* convb_b   = (const float*)d_in[17];
    const float* xp_w      = (const float*)d_in[18];
    const float* xpb_w     = (const float*)d_in[19];
    const float* dtp_w     = (const float*)d_in[20];
    const float* dtp_b     = (const float*)d_in[21];
    const float* dtpb_w    = (const float*)d_in[22];
    const float* dtpb_b    = (const float*)d_in[23];
    const float* A_log     = (const float*)d_in[24];
    const float* Ab_log    = (const float*)d_in[25];
    const float* Dskip     = (const float*)d_in[26];
    const float* Dskipb    = (const float*)d_in[27];
    const float* out_pw    = (const float*)d_in[28];
    const float* gamma     = (const float*)d_in[29];
    const float* ts_w      = (const float*)d_in[30];
    const float* ts_b      = (const float*)d_in[31];
    const float* ln2_g     = (const float*)d_in[32];
    const float* ln2_b     = (const float*)d_in[33];
    const float* mlp_w1    = (const float*)d_in[34];
    const float* mlp_b1    = (const float*)d_in[35];
    const float* mlp_w2    = (const float*)d_in[36];
    const float* mlp_b2    = (const float*)d_in[37];
    float* out = (float*)d_out;

    // Workspace layout (floats). xc aliases [xn|hU], hidden aliases xz,
    // x2 aliases xm, lnf aliases hV -- all safe by kernel ordering.
    float* ws = (float*)d_ws;
    const size_t S = (size_t)MM * DD;      // one M x 256 plane
    float* xn    = ws + 0 * S;
    float* hU    = ws + 1 * S;
    float* xc    = ws + 0 * S;             // M x 512, overlays xn|hU
    float* xm    = ws + 2 * S;
    float* x2    = ws + 2 * S;             // overlays xm
    float* hV    = ws + 3 * S;
    float* lnf   = ws + 3 * S;             // overlays hV
    float* x1    = ws + 4 * S;
    float* xf    = ws + 5 * S;
    float* xz    = ws + 6 * S;             // M x 1024 (u | z)
    float* hidden= ws + 6 * S;             // overlays xz
    float* dtb   = ws + 10 * S;            // M x 512
    float* ybuf  = ws + 12 * S;            // M x 512
    float* xdbl  = ws + 14 * S;            // M x 32

    const dim3 blk(256);
    const int  rowsBlk = MM / 8;           // wave-per-row kernels: 4352
    const int  gx = MM / (16 * 8);         // 272 gemm m-blocks

    // 1) xn = LN(x)*g_gcn+b_gcn ; xm = LN(x)*g1+b1  (shared stats)
    ln_dual_kernel<<<rowsBlk, blk, 0, stream>>>(x, ln_gcn_g, ln_gcn_b, ln1_g, ln1_b, xn, xm);

    // 2) hV = xn @ Vw^T + Vb ; hU = xn @ Uw^T + Ub
    gemm_f32_wmma<8, 32, 1><<<dim3(gx, 2), blk, 0, stream>>>(xn, gcn_V_w, gcn_V_b, nullptr, nullptr, hV, MM, DD, DD, DD);
    gemm_f32_wmma<8, 32, 1><<<dim3(gx, 2), blk, 0, stream>>>(xn, gcn_U_w, gcn_U_b, nullptr, nullptr, hU, MM, DD, DD, DD);

    // 3) x1 = x + relu(xn + BN(adj-agg(hV) + hU))
    gcn_fuse_kernel<<<(MM * DD) / 256, blk, 0, stream>>>(x, xn, hV, hU, bn_g, bn_b, bn_mean, bn_var, x1);

    // 4) xz = xm @ in_proj_w^T   (u | z)
    gemm_f32_wmma<8, 32, 0><<<dim3(gx, 8), blk, 0, stream>>>(xm, in_proj_w, nullptr, nullptr, nullptr, xz, MM, 2 * EE, DD, DD);

    // 5) forward SSM branch
    conv_silu_kernel<<<(MM * EE) / 256, blk, 0, stream>>>(xz, conv_w, conv_b, xc, 0);
    gemm_f32_wmma<2, 32, 0><<<dim3(gx, 1), blk, 0, stream>>>(xc, xp_w, nullptr, nullptr, nullptr, xdbl, MM, RR + 2 * NN, EE, EE);
    gemm_f32_wmma<8, 16, 2><<<dim3(gx, 4), blk, 0, stream>>>(xdbl, dtp_w, dtp_b, nullptr, nullptr, dtb, MM, EE, RR, RR + 2 * NN);
    ssm_scan_kernel<<<(BB * EE) / 256, blk, 0, stream>>>(dtb, xc, xdbl, xz, A_log, Dskip, ybuf, 0, 0);

    // 6) backward SSM branch (time-reversed), fused combine into ybuf
    conv_silu_kernel<<<(MM * EE) / 256, blk, 0, stream>>>(xz, convb_w, convb_b, xc, 1);
    gemm_f32_wmma<2, 32, 0><<<dim3(gx, 1), blk, 0, stream>>>(xc, xpb_w, nullptr, nullptr, nullptr, xdbl, MM, RR + 2 * NN, EE, EE);
    gemm_f32_wmma<8, 16, 2><<<dim3(gx, 4), blk, 0, stream>>>(xdbl, dtpb_w, dtpb_b, nullptr, nullptr, dtb, MM, EE, RR, RR + 2 * NN);
    ssm_scan_kernel<<<(BB * EE) / 256, blk, 0, stream>>>(dtb, xc, xdbl, xz, Ab_log, Dskipb, ybuf, 1, 1);

    // 7) x2 = x + (y @ out_proj_w^T) * gamma
    gemm_f32_wmma<8, 32, 4><<<dim3(gx, 2), blk, 0, stream>>>(ybuf, out_pw, nullptr, x, gamma, x2, MM, DD, EE, EE);

    // 8) softmax token fusion -> xf
    ts_fuse_kernel<<<rowsBlk, blk, 0, stream>>>(x1, x2, ts_w, ts_b, xf);

    // 9) MLP: out = xf + gelu(LN(xf)@W1^T+b1) @ W2^T + b2
    ln_kernel<<<rowsBlk, blk, 0, stream>>>(xf, ln2_g, ln2_b, lnf);
    gemm_f32_wmma<8, 32, 3><<<dim3(gx, 8), blk, 0, stream>>>(lnf, mlp_w1, mlp_b1, nullptr, nullptr, hidden, MM, HH, DD, DD);
    gemm_f32_wmma<8, 32, 5><<<dim3(gx, 2), blk, 0, stream>>>(hidden, mlp_w2, mlp_b2, xf, nullptr, out, MM, DD, HH, HH);
}